// MeshTorchLayer_56160992363153
// MI455X (gfx1250) — compile-verified
//
#include <hip/hip_runtime.h>
#include <hip/hip_bf16.h>

#define UU 512
#define LL 512
#define BB 256
static __device__ constexpr float PI_F = 3.14159265358979323846f;

typedef float v2f __attribute__((ext_vector_type(2)));
typedef float v8f __attribute__((ext_vector_type(8)));

// ---------------------------------------------------------------------------
// Kernel 1: precompute diag/off coefficients, shape (L, U) complex each.
// diagw[l*U+u] = diag[:, l, u] (post-transpose layout of the reference).
// ---------------------------------------------------------------------------
__global__ __launch_bounds__(256) void coeff_kernel(
    const float* __restrict__ theta, const float* __restrict__ phi,
    const float* __restrict__ mask,  const float* __restrict__ enn,
    const float* __restrict__ enp,   const float* __restrict__ epn,
    const float* __restrict__ epp,
    float2* __restrict__ diagw, float2* __restrict__ offw) {
  int idx = blockIdx.x * blockDim.x + threadIdx.x;   // [0, L*U)
  int l = idx / UU;
  int u = idx % UU;

  // phase-shifter complex exp at column uu (odd columns carry angle 0)
  auto psl = [&](const float* ang, int uu) -> float2 {
    uu &= (UU - 1);
    if (uu & 1) return make_float2(1.0f, 0.0f);
    int k = uu >> 1;
    float m = mask[l * (UU / 2) + k];
    float a = ang[l * (UU / 2) + k] * m + (1.0f - m) * PI_F;
    float s, c;
    __sincosf(a, &s, &c);
    return make_float2(c, s);
  };

  float2 ip_m1 = psl(theta, u - 1);
  float2 ip_0  = psl(theta, u);
  float2 ip_p1 = psl(theta, u + 1);
  float2 ep_m1 = psl(phi,   u - 1);
  float2 ep_0  = psl(phi,   u);

  int um1 = (u - 1) & (UU - 1);
  float enn_u = enn[u * LL + l],  enn_m = enn[um1 * LL + l];
  float epp_u = epp[u * LL + l],  epp_m = epp[um1 * LL + l];
  float enp_u = enp[u * LL + l],  enp_m = enp[um1 * LL + l];
  float epn_u = epn[u * LL + l],  epn_m = epn[um1 * LL + l];

  // s11[u] + s22[u] (componentwise real scaling of complex ipsl)
  float sdr = epp_u * ip_0.x - enn_u * ip_p1.x - enn_m * ip_m1.x + epp_m * ip_0.x;
  float sdi = epp_u * ip_0.y - enn_u * ip_p1.y - enn_m * ip_m1.y + epp_m * ip_0.y;

  // s21[u] + s12[u] = i * ( Cv(u) + Bv(u-1) )
  float cvr = epn_u * ip_0.x + enp_u * ip_p1.x + enp_m * ip_m1.x + epn_m * ip_0.x;
  float cvi = epn_u * ip_0.y + enp_u * ip_p1.y + enp_m * ip_m1.y + epn_m * ip_0.y;
  float sor = -cvi, soi = cvr;

  // diag = 0.5 * epsl(u) * (s11+s22);  off = 0.5 * epsl(u-1) * (s21+s12)
  diagw[idx] = make_float2(0.5f * (ep_0.x * sdr - ep_0.y * sdi),
                           0.5f * (ep_0.x * sdi + ep_0.y * sdr));
  offw[idx]  = make_float2(0.5f * (ep_m1.x * sor - ep_m1.y * soi),
                           0.5f * (ep_m1.x * soi + ep_m1.y * sor));
}

// ---------------------------------------------------------------------------
// Kernel 2: invert perms[1..L]: invp[l][c] = u  s.t.  perms[l+1][u] == c
// (lets the scan fold the roll into the LDS write-back scatter)
// ---------------------------------------------------------------------------
__global__ __launch_bounds__(256) void invp_kernel(const int* __restrict__ perms,
                                                   int* __restrict__ invp) {
  int idx = blockIdx.x * blockDim.x + threadIdx.x;   // [0, L*U)
  int l = idx / UU;
  int u = idx % UU;
  int c = perms[(l + 1) * UU + u];
  invp[l * UU + c] = u;
}

// ---------------------------------------------------------------------------
// Kernel 3: fused 512-layer scan. One WG = 16 batch rows x full U in LDS.
// Pair mixing done as block-diagonal 16x16 real GEMM on the WMMA f32 pipe:
// D(16x16) = sum_j A_j(16x4) x B_j(4x16), 4 chained v_wmma_f32_16x16x4_f32.
// ---------------------------------------------------------------------------
__global__ __launch_bounds__(256) void scan_kernel(
    const float* __restrict__ x, const float* __restrict__ gamma,
    const int* __restrict__ perms, const float2* __restrict__ diagw,
    const float2* __restrict__ offw, const int* __restrict__ invp,
    float* __restrict__ out) {
  __shared__ float2 S[16 * UU];   // 64 KB state: 16 batch rows x 512 cols, complex

  const int tid  = threadIdx.x;
  const int b0   = blockIdx.x * 16;
  const int wave = tid >> 5;
  const int lane = tid & 31;
  const int n    = lane & 15;   // batch column within the 16-wide tile
  const int hi   = lane >> 4;   // 0: K/M cols 0-1 half, 1: 2-3 half
  const int m    = lane & 15;   // A-operand row (M)
  const int r    = m & 3;       // row within 4x4 pair block
  const int blk  = m >> 2;      // which pair block this lane's row belongs to

  // ---- init: state = cc_mul(x, exp(i*gamma)) gathered through perms[0] ----
  for (int e = 0; e < (16 * UU) / 256; ++e) {
    int idx = tid + e * 256;
    int rr = idx >> 9;            // batch row 0..15
    int u  = idx & (UU - 1);
    int p  = perms[u];            // perms row 0
    float xr = x[(size_t)(b0 + rr) * UU + p];
    float xi = x[(size_t)BB * UU + (size_t)(b0 + rr) * UU + p];
    float sg, cg;
    __sincosf(gamma[p], &sg, &cg);
    S[rr * UU + u] = make_float2(xr * cg - xi * sg, xr * sg + xi * cg);
  }
  __syncthreads();

  for (int l = 0; l < LL; ++l) {
    const float2* dl = diagw + (size_t)l * UU;
    const float2* ol = offw  + (size_t)l * UU;
    const int*    il = invp  + (size_t)l * UU;
    // pull next layer's coefficient rows toward L2/L0 while we compute
    __builtin_prefetch(dl + UU, 0, 0);
    __builtin_prefetch(ol + UU, 0, 0);
    __builtin_prefetch(il + UU, 0, 0);

    v8f acc[8];
#pragma unroll
    for (int t = 0; t < 8; ++t) {
      const int cb = (wave * 8 + t) * 8;   // 8-column tile base
      v8f c = {0.f, 0.f, 0.f, 0.f, 0.f, 0.f, 0.f, 0.f};
#pragma unroll
      for (int j = 0; j < 4; ++j) {
        const int col = cb + 2 * j + hi;
        // B_j operand: (re,im) of pair-j column `col`, batch row n.
        float2 sb = S[n * UU + col];
        v2f bop;  bop[0] = sb.x;  bop[1] = sb.y;
        // A_j operand: 4x4 complex-as-real pair block, zero outside block j.
        // rows 0/1 use diag(c0),off(c1); rows 2/3 use off(c0),diag(c1).
        bool use_diag = ((r < 2) == (hi == 0));
        float2 p = use_diag ? dl[col] : ol[col];
        float a0, a1;
        if (r & 1) { a0 = p.y;  a1 = p.x;  }   // odd rows:  ( im,  re)
        else       { a0 = p.x;  a1 = -p.y; }   // even rows: ( re, -im)
        if (blk != j) { a0 = 0.f; a1 = 0.f; }
        v2f aop;  aop[0] = a0;  aop[1] = a1;
        c = __builtin_amdgcn_wmma_f32_16x16x4_f32(
                false, aop, false, bop, (short)0, c, false, false);
      }
      acc[t] = c;
    }
    __syncthreads();   // everyone done reading S
    // scatter D back through the inverse roll permutation
#pragma unroll
    for (int t = 0; t < 8; ++t) {
      const int cb = (wave * 8 + t) * 8;
#pragma unroll
      for (int v = 0; v < 8; v += 2) {
        int mm = v + hi * 8;           // D row (even -> re, odd -> im)
        int cc = cb + (mm >> 1);       // source column of this value pair
        int uo = il[cc];               // destination column after the roll
        S[n * UU + uo] = make_float2(acc[t][v], acc[t][v + 1]);
      }
    }
    __syncthreads();   // writes visible before next layer's reads
  }

  // ---- write out (2, B, U) ----
  for (int e = 0; e < (16 * UU) / 256; ++e) {
    int idx = tid + e * 256;
    int rr = idx >> 9;
    int u  = idx & (UU - 1);
    float2 s = S[rr * UU + u];
    out[(size_t)(b0 + rr) * UU + u] = s.x;
    out[(size_t)BB * UU + (size_t)(b0 + rr) * UU + u] = s.y;
  }
}

// ---------------------------------------------------------------------------
extern "C" void kernel_launch(void* const* d_in, const int* in_sizes, int n_in,
                              void* d_out, int out_size, void* d_ws, size_t ws_size,
                              hipStream_t stream) {
  const float* x     = (const float*)d_in[0];
  const float* theta = (const float*)d_in[1];
  const float* phi   = (const float*)d_in[2];
  const float* gamma = (const float*)d_in[3];
  const float* mask  = (const float*)d_in[4];
  const float* enn   = (const float*)d_in[5];
  const float* enp   = (const float*)d_in[6];
  const float* epn   = (const float*)d_in[7];
  const float* epp   = (const float*)d_in[8];
  const int*   perms = (const int*)d_in[9];
  // d_in[10] = pairwise_perm (adjacent swap by construction; folded into the
  // 2x2 pair-block structure of the WMMA A operand)

  char* ws = (char*)d_ws;
  float2* diagw = (float2*)(ws);                               // 2 MB
  float2* offw  = (float2*)(ws + (size_t)LL * UU * 8);         // 2 MB
  int*    invp  = (int*)   (ws + (size_t)LL * UU * 16);        // 1 MB

  const int threads = 256;
  const int blocksLU = (LL * UU) / threads;                    // 1024

  coeff_kernel<<<blocksLU, threads, 0, stream>>>(theta, phi, mask, enn, enp,
                                                 epn, epp, diagw, offw);
  invp_kernel<<<blocksLU, threads, 0, stream>>>(perms, invp);
  scan_kernel<<<BB / 16, threads, 0, stream>>>(x, gamma, perms, diagw, offw,
                                               invp, (float*)d_out);
}